// aspect_rating_2_39900246180589
// MI455X (gfx1250) — compile-verified
//
#include <hip/hip_runtime.h>
#include <hip/hip_bf16.h>

// Problem constants (from the reference)
#define R_    30000
#define L_    40
#define D_    200
#define A_    32
#define BSZ_  10000
#define NNZ_  300000

typedef __attribute__((ext_vector_type(2))) float v2f;
typedef __attribute__((ext_vector_type(8))) float v8f;

// ---------------------------------------------------------------------------
// Kernel 1: Q = review_positive @ W_M   ([R,D] = [R,D] @ [D,D])
// One wave per 16-row strip of Q; the wave holds all 13 column tiles in
// registers (13 x v8f accumulators) and steps K by 4 with
// V_WMMA_F32_16X16X4_F32. A is loaded once per k-step and reused by 13
// back-to-back independent WMMAs (disjoint accumulators -> no D->A/B hazard).
// Out-of-range columns (200..207 of tile 12) load a CLAMPED address instead
// of being masked: a WMMA output column depends only on the matching B
// column, and those columns are never stored -> unconditional clean loads,
// no exec-mask churn.
// VGPR layouts (ISA 7.12.2):
//   A (16x4, MxK): lane = M(0..15) + 16*half, vgpr j -> K = j + 2*half
//   B (4x16, KxN): lane = N(0..15) + 16*half, vgpr j -> K = j + 2*half
//   C/D (16x16):   vgpr v, half h -> M = v + 8h, N = lane&15
// ---------------------------------------------------------------------------
#define NT_ 13   // ceil(200/16) column tiles

__global__ __launch_bounds__(32) void q_gemm_kernel(const float* __restrict__ rp,
                                                    const float* __restrict__ wm,
                                                    float* __restrict__ q) {
    const int lane = threadIdx.x;
    const int row0 = blockIdx.x * 16;       // over R (30000/16 = 1875 exact)
    const int half = lane >> 4;
    const int lm   = lane & 15;
    const int arow = (row0 + lm) * D_;

    v8f acc[NT_];
    const v8f zero = {};
#pragma unroll
    for (int t = 0; t < NT_; ++t) acc[t] = zero;

    for (int k = 0; k < D_; k += 4) {
        const int ka = k + 2 * half;
        v2f a;
        a.x = rp[arow + ka + 0];
        a.y = rp[arow + ka + 1];
#pragma unroll
        for (int t = 0; t < NT_; ++t) {
            int col = t * 16 + lm;
            if (col > D_ - 1) col = D_ - 1;   // clamp: dead columns, never stored
            v2f b;
            b.x = wm[(ka + 0) * D_ + col];
            b.y = wm[(ka + 1) * D_ + col];
            // (neg_a, A, neg_b, B, c_mod, C, reuse_a, reuse_b)
            acc[t] = __builtin_amdgcn_wmma_f32_16x16x4_f32(false, a, false, b,
                                                           (short)0, acc[t],
                                                           false, false);
        }
    }

#pragma unroll
    for (int t = 0; t < NT_; ++t) {
        const int col = t * 16 + lm;
        if (col < D_) {
#pragma unroll
            for (int v = 0; v < 8; ++v) {
                q[(row0 + v + 8 * half) * D_ + col] = acc[t][v];
            }
        }
    }
}

// ---------------------------------------------------------------------------
// Kernel 2: fused attention per review. One wave32 per review, 8 reviews/block.
//   dx[l]  = emb[tok[l],:] . Q[r,:]         (lane-per-word, float4 loads)
//   ax     = softmax(dx) over L=40          (wave shfl reductions)
//   z[d]   = sum_l ax[l]*emb[tok[l],d]      (lane-per-d chunks, LDS stage)
//   p_t[a] = b_W[a] + z . W_W[a,:]          (lane-per-aspect, A==32==wave32)
// Embedding table (40 MB) is L2-resident; re-reads are cheap.
// ---------------------------------------------------------------------------
__global__ __launch_bounds__(256) void attention_kernel(
        const int*   __restrict__ hist,   // [R, L]
        const float* __restrict__ q,      // [R, D]
        const float* __restrict__ emb,    // [VOCAB, D]
        const float* __restrict__ ww,     // [A, D]
        const float* __restrict__ bw,     // [A]
        float*       __restrict__ pt)     // [R, A]
{
    __shared__ int   toks[8][L_];
    __shared__ float axs [8][L_];
    __shared__ __align__(16) float zs[8][D_];

    const int lane = threadIdx.x & 31;
    const int wave = threadIdx.x >> 5;
    const int r    = blockIdx.x * 8 + wave;     // 30000/8 = 3750 exact

    // load token ids: l = lane (0..31) and l = 32+lane (lane<8)
    const int tokA = hist[r * L_ + lane];
    const int tokB = (lane < 8) ? hist[r * L_ + 32 + lane] : 0;
    toks[wave][lane] = tokA;
    if (lane < 8) toks[wave][32 + lane] = tokB;

    // phase 1: dx[l] = emb_row(tok[l]) . q[r,:]   (b128 loads, rows are 800B)
    const float4* qr4 = reinterpret_cast<const float4*>(q + r * D_);
    const float4* ea4 = reinterpret_cast<const float4*>(emb + tokA * D_);
    const float4* eb4 = reinterpret_cast<const float4*>(emb + tokB * D_);
    float dxa = 0.f, dxb = 0.f;
    for (int i = 0; i < D_ / 4; ++i) {
        const float4 qv = qr4[i];          // uniform-within-wave broadcast load
        const float4 av = ea4[i];
        const float4 bv = eb4[i];
        dxa = fmaf(av.x, qv.x, fmaf(av.y, qv.y, fmaf(av.z, qv.z, fmaf(av.w, qv.w, dxa))));
        dxb = fmaf(bv.x, qv.x, fmaf(bv.y, qv.y, fmaf(bv.z, qv.z, fmaf(bv.w, qv.w, dxb))));
    }

    // softmax over 40 values (dxa: all 32 lanes, dxb: lanes 0..7)
    float m = fmaxf(dxa, (lane < 8) ? dxb : -1e30f);
    for (int off = 16; off > 0; off >>= 1) m = fmaxf(m, __shfl_xor(m, off, 32));
    const float expa = __expf(dxa - m);
    const float expb = (lane < 8) ? __expf(dxb - m) : 0.f;
    float s = expa + expb;
    for (int off = 16; off > 0; off >>= 1) s += __shfl_xor(s, off, 32);
    const float inv = 1.0f / s;
    axs[wave][lane] = expa * inv;
    if (lane < 8) axs[wave][32 + lane] = expb * inv;
    __syncthreads();

    // phase 2: z[d] = sum_l ax[l] * emb[tok[l], d]  (lane-coalesced over d)
    for (int c = 0; c < 7; ++c) {
        const int d = c * 32 + lane;
        if (d < D_) {
            float acc = 0.f;
#pragma unroll 8
            for (int l = 0; l < L_; ++l) {
                acc = fmaf(axs[wave][l], emb[toks[wave][l] * D_ + d], acc);
            }
            zs[wave][d] = acc;
        }
    }
    __syncthreads();

    // phase 3: p_t[r, lane] = b_W[lane] + z . W_W[lane, :]  (b128 loads)
    const float4* z4 = reinterpret_cast<const float4*>(&zs[wave][0]);
    const float4* w4 = reinterpret_cast<const float4*>(ww + lane * D_);
    float p = bw[lane];
    for (int i = 0; i < D_ / 4; ++i) {
        const float4 zv = z4[i];
        const float4 wv = w4[i];
        p = fmaf(zv.x, wv.x, fmaf(zv.y, wv.y, fmaf(zv.z, wv.z, fmaf(zv.w, wv.w, p))));
    }
    pt[r * A_ + lane] = p;
}

// ---------------------------------------------------------------------------
// Kernel 3: sparse scatter  out[row,:] += val * p_t[col,:]
// One wave per nonzero, lane == aspect (A=32). gridDim.y selects user/item.
// ---------------------------------------------------------------------------
__global__ __launch_bounds__(256) void spmm_scatter_kernel(
        const int* __restrict__ idx_u, const float* __restrict__ val_u,
        const int* __restrict__ idx_i, const float* __restrict__ val_i,
        const float* __restrict__ pt,
        float*       __restrict__ out)   // [2, B, A] flat
{
    const int lane = threadIdx.x & 31;
    const int nz   = (int)((blockIdx.x * (unsigned)blockDim.x + threadIdx.x) >> 5);
    if (nz >= NNZ_) return;
    const int*   idx = (blockIdx.y == 0) ? idx_u : idx_i;
    const float* val = (blockIdx.y == 0) ? val_u : val_i;
    float* o = out + (size_t)blockIdx.y * BSZ_ * A_;

    const int   row = idx[nz];            // index[0, nz]
    const int   col = idx[NNZ_ + nz];     // index[1, nz]
    const float v   = val[nz];
    atomicAdd(&o[row * A_ + lane], v * pt[col * A_ + lane]);
}

__global__ void zero_kernel(float* __restrict__ p, int n) {
    const int i = blockIdx.x * blockDim.x + threadIdx.x;
    if (i < n) p[i] = 0.f;
}

// ---------------------------------------------------------------------------
extern "C" void kernel_launch(void* const* d_in, const int* in_sizes, int n_in,
                              void* d_out, int out_size, void* d_ws, size_t ws_size,
                              hipStream_t stream) {
    const int*   hist = (const int*)  d_in[0];   // historical_review [R,L]
    const float* rp   = (const float*)d_in[1];   // review_positive  [R,D]
    // d_in[2] review_negative: unused by the reference forward pass
    const int*   uidx = (const int*)  d_in[3];   // user_histor_index [2,NNZ]
    const float* uval = (const float*)d_in[4];   // user_histor_value [NNZ]
    const int*   iidx = (const int*)  d_in[5];   // item_histor_index [2,NNZ]
    const float* ival = (const float*)d_in[6];   // item_histor_value [NNZ]
    // d_in[7] batch_size scalar (compile-time BSZ_)
    const float* emb  = (const float*)d_in[8];   // emb_table [VOCAB,D]
    const float* wm   = (const float*)d_in[9];   // W_M [D,D]
    const float* ww   = (const float*)d_in[10];  // W_W [A,D]
    const float* bwv  = (const float*)d_in[11];  // b_W [A]
    float* out = (float*)d_out;                  // [2, B, A]

    // workspace layout: Q [R*D] f32, then p_t [R*A] f32  (~27.8 MB)
    float* q  = (float*)d_ws;
    float* pt = q + (size_t)R_ * D_;

    // 1) zero scatter target
    {
        const int n = 2 * BSZ_ * A_;
        zero_kernel<<<(n + 255) / 256, 256, 0, stream>>>(out, n);
    }
    // 2) Q = RP @ W_M via f32 WMMA (wave per 16-row strip, 650 WMMAs/wave)
    q_gemm_kernel<<<R_ / 16, 32, 0, stream>>>(rp, wm, q);
    // 3) fused attention + p_t projection
    attention_kernel<<<R_ / 8, 256, 0, stream>>>(hist, q, emb, ww, bwv, pt);
    // 4) both sparse scatters
    {
        dim3 grid((NNZ_ * 32 + 255) / 256, 2);
        spmm_scatter_kernel<<<grid, 256, 0, stream>>>(uidx, uval, iidx, ival, pt, out);
    }
}